// BoxGauss_1288490188936
// MI455X (gfx1250) — compile-verified
//
#include <hip/hip_runtime.h>
#include <hip/hip_bf16.h>
#include <math.h>

// MI455X / gfx1250, wave32. Bandwidth-bound masked-MSE loss.
// ~183 MB streamed @ 23.3 TB/s ~= 8 us ideal. Reduction summation done with
// V_WMMA_F32_16X16X4_F32 (B = ones) for full-f32 tensor-pipe accumulation.

typedef float v2f __attribute__((ext_vector_type(2)));
typedef float v8f __attribute__((ext_vector_type(8)));

#define NBOX 256

// ---------------------------------------------------------------------------
// Kernel 1: per-level Gaussian mask m[b,y,x] + per-block partial sum of m.
// Grid is an exact multiple of 256 pixels; every thread owns one pixel.
// ---------------------------------------------------------------------------
__global__ __launch_bounds__(256) void mask_kernel(
    const float* __restrict__ bboxes, const int* __restrict__ batch_idx,
    int nbox, int S, float* __restrict__ m, float* __restrict__ partial)
{
    __shared__ float s_xc[NBOX], s_yc[NBOX], s_iw[NBOX], s_ih[NBOX];
    __shared__ int   s_xl[NBOX], s_xr[NBOX], s_yt[NBOX], s_yd[NBOX], s_b[NBOX];

    const int tid = threadIdx.x;
    if (tid < NBOX) {
        if (tid < nbox) {
            float bx = bboxes[tid * 4 + 0];
            float by = bboxes[tid * 4 + 1];
            float bw = bboxes[tid * 4 + 2];
            float bh = bboxes[tid * 4 + 3];
            int xc = (int)floorf(bx * (float)S);
            int yc = (int)floorf(by * (float)S);
            int w  = (int)floorf(bw * (float)S);
            int h  = (int)floorf(bh * (float)S);
            int xl = max(xc - w / 2, 0);
            int xr = min(xc + w / 2, S - 1);
            int yt = max(yc - h / 2, 0);
            int yd = min(yc + h / 2, S - 1);
            float width  = (float)(xr - xl + 1);
            float height = (float)(yd - yt + 1);
            s_xc[tid] = (float)xc;  s_yc[tid] = (float)yc;
            // STD=2: STD^2*(width/2)^2 == width^2
            s_iw[tid] = 1.0f / (width * width);
            s_ih[tid] = 1.0f / (height * height);
            s_xl[tid] = xl; s_xr[tid] = xr; s_yt[tid] = yt; s_yd[tid] = yd;
            s_b[tid]  = batch_idx[tid];
        } else {
            s_b[tid] = -1;
        }
    }
    __syncthreads();

    const int idx = blockIdx.x * 256 + tid;
    const int SS  = S * S;
    const int b   = idx / SS;
    const int pix = idx % SS;
    const int y   = pix / S;
    const int x   = pix % S;
    const float fx = (float)x, fy = (float)y;

    float v = 0.0f;   // segment_max floored at 0 (empty segments -> 0)
    #pragma unroll 4
    for (int n = 0; n < NBOX; ++n) {
        bool hit = (s_b[n] == b) & (x >= s_xl[n]) & (x <= s_xr[n]) &
                   (y >= s_yt[n]) & (y <= s_yd[n]);
        if (hit) {
            float dx = fx - s_xc[n];
            float dy = fy - s_yc[n];
            float g = expf(-(dx * dx * s_iw[n] + dy * dy * s_ih[n]));
            v = fmaxf(v, g);
        }
    }
    m[idx] = v;

    // deterministic block reduction of mask sum
    __shared__ float red[256];
    red[tid] = v;
    __syncthreads();
    #pragma unroll
    for (int s2 = 128; s2 > 0; s2 >>= 1) {
        if (tid < s2) red[tid] += red[tid + s2];
        __syncthreads();
    }
    if (tid == 0) partial[blockIdx.x] = red[0];
}

// ---------------------------------------------------------------------------
// Kernel 2: sum over all elements of (m[b,y,x]*(p-t))^2, float4-vectorized.
// Summation via V_WMMA_F32_16X16X4_F32 with B = ones (row-sum accumulate),
// two independent accumulator chains to hide WMMA latency. EXEC is all-ones
// by construction (exact grid coverage, no divergence).
// ---------------------------------------------------------------------------
__global__ __launch_bounds__(256) void sumsq_kernel(
    const float4* __restrict__ p, const float4* __restrict__ t,
    const float4* __restrict__ m, float* __restrict__ partial,
    int CSS4, int SS4, int iters, int stride)
{
    v8f c0 = {};
    v8f c1 = {};
    v2f ones; ones[0] = 1.0f; ones[1] = 1.0f;

    const int base = blockIdx.x * 256 + threadIdx.x;
    for (int k = 0; k < iters; ++k) {
        int i4 = base + k * stride;
        float4 pv = p[i4];
        float4 tv = t[i4];
        int b    = i4 / CSS4;        // image index
        int pix4 = i4 % SS4;         // float4 pixel offset within image plane
        float4 mv = m[b * SS4 + pix4];

        float d0 = mv.x * (pv.x - tv.x);
        float d1 = mv.y * (pv.y - tv.y);
        float d2 = mv.z * (pv.z - tv.z);
        float d3 = mv.w * (pv.w - tv.w);

        v2f a01; a01[0] = d0 * d0; a01[1] = d1 * d1;
        v2f a23; a23[0] = d2 * d2; a23[1] = d3 * d3;

        // D = A x ones(4x16) + C : accumulates 64 f32 values per instruction
        c0 = __builtin_amdgcn_wmma_f32_16x16x4_f32(
                 false, a01, false, ones, (short)0, c0, false, false);
        c1 = __builtin_amdgcn_wmma_f32_16x16x4_f32(
                 false, a23, false, ones, (short)0, c1, false, false);
    }

    // Sum of D over all lanes & regs == 16 * (total fed), since the 16
    // columns are identical row-sum copies.
    float s = 0.0f;
    #pragma unroll
    for (int r = 0; r < 8; ++r) s += c0[r] + c1[r];
    #pragma unroll
    for (int off = 16; off > 0; off >>= 1) s += __shfl_xor(s, off, 32);
    s *= (1.0f / 16.0f);   // undo 16 duplicate columns

    __shared__ float wred[8];
    const int lane = threadIdx.x & 31;
    const int wid  = threadIdx.x >> 5;
    if (lane == 0) wred[wid] = s;
    __syncthreads();
    if (threadIdx.x == 0) {
        float tot = 0.0f;
        #pragma unroll
        for (int w = 0; w < 8; ++w) tot += wred[w];
        partial[blockIdx.x] = tot;
    }
}

// ---------------------------------------------------------------------------
// Kernel 3: reduce all partial arrays (fixed order -> deterministic) and
// combine: total = (1/3) * sum_l  sq_l / (C_l * maskSum_l)
// ---------------------------------------------------------------------------
__global__ __launch_bounds__(256) void finalize_kernel(
    const float* __restrict__ ws, int mp0o, int mp0n, int mp1o, int mp1n,
    int mp2o, int mp2n, int sp0o, int sp0n, int sp1o, int sp1n,
    int sp2o, int sp2n, float* __restrict__ out)
{
    __shared__ float red[256];
    float results[6];
    const int offs[6] = {mp0o, mp1o, mp2o, sp0o, sp1o, sp2o};
    const int lens[6] = {mp0n, mp1n, mp2n, sp0n, sp1n, sp2n};

    for (int a = 0; a < 6; ++a) {
        float loc = 0.0f;
        for (int i = threadIdx.x; i < lens[a]; i += 256) loc += ws[offs[a] + i];
        red[threadIdx.x] = loc;
        __syncthreads();
        #pragma unroll
        for (int s2 = 128; s2 > 0; s2 >>= 1) {
            if (threadIdx.x < s2) red[threadIdx.x] += red[threadIdx.x + s2];
            __syncthreads();
        }
        results[a] = red[0];
        __syncthreads();
    }

    if (threadIdx.x == 0) {
        float total = results[3] / (128.0f * results[0])
                    + results[4] / (256.0f * results[1])
                    + results[5] / (512.0f * results[2]);
        out[0] = total * (1.0f / 3.0f);
    }
}

// ---------------------------------------------------------------------------
extern "C" void kernel_launch(void* const* d_in, const int* in_sizes, int n_in,
                              void* d_out, int out_size, void* d_ws, size_t ws_size,
                              hipStream_t stream) {
    const float* p0 = (const float*)d_in[0];   // [16,128,80,80]
    const float* p1 = (const float*)d_in[1];   // [16,256,40,40]
    const float* p2 = (const float*)d_in[2];   // [16,512,20,20]
    const float* t0 = (const float*)d_in[3];
    const float* t1 = (const float*)d_in[4];
    const float* t2 = (const float*)d_in[5];
    const float* bboxes = (const float*)d_in[6];
    const int*   bidx   = (const int*)d_in[8];
    const int nbox = in_sizes[6] / 4;          // 256

    float* ws  = (float*)d_ws;
    float* out = (float*)d_out;

    // workspace layout (floats)
    const int m0o = 0;                      // 16*80*80 = 102400
    const int m1o = 102400;                 // 16*40*40 =  25600
    const int m2o = 128000;                 // 16*20*20 =   6400
    const int mp0o = 134400, mp0n = 400;    // mask partials
    const int mp1o = 134800, mp1n = 100;
    const int mp2o = 134900, mp2n = 25;
    const int sp0o = 134928, sp0n = 1280;   // sumsq partials (16B aligned)
    const int sp1o = 136208, sp1n = 1280;
    const int sp2o = 137488, sp2n = 640;    // total ~552 KB of ws

    // masks (tiny) — exact grids: 16*S*S / 256 blocks
    mask_kernel<<<400, 256, 0, stream>>>(bboxes, bidx, nbox, 80, ws + m0o, ws + mp0o);
    mask_kernel<<<100, 256, 0, stream>>>(bboxes, bidx, nbox, 40, ws + m1o, ws + mp1o);
    mask_kernel<<< 25, 256, 0, stream>>>(bboxes, bidx, nbox, 20, ws + m2o, ws + mp2o);

    // streaming reductions — exact coverage, no tails
    // L0: 3,276,800 f4 = 1280*256*10 ; CSS4 = 128*6400/4 = 204800 ; SS4 = 1600
    sumsq_kernel<<<1280, 256, 0, stream>>>((const float4*)p0, (const float4*)t0,
        (const float4*)(ws + m0o), ws + sp0o, 204800, 1600, 10, 1280 * 256);
    // L1: 1,638,400 f4 = 1280*256*5 ; CSS4 = 256*1600/4 = 102400 ; SS4 = 400
    sumsq_kernel<<<1280, 256, 0, stream>>>((const float4*)p1, (const float4*)t1,
        (const float4*)(ws + m1o), ws + sp1o, 102400, 400, 5, 1280 * 256);
    // L2:   819,200 f4 =  640*256*5 ; CSS4 = 512*400/4 = 51200 ; SS4 = 100
    sumsq_kernel<<< 640, 256, 0, stream>>>((const float4*)p2, (const float4*)t2,
        (const float4*)(ws + m2o), ws + sp2o, 51200, 100, 5, 640 * 256);

    finalize_kernel<<<1, 256, 0, stream>>>(ws, mp0o, mp0n, mp1o, mp1n, mp2o, mp2n,
                                           sp0o, sp0n, sp1o, sp1n, sp2o, sp2n, out);
}